// Decoder_64819646431327
// MI455X (gfx1250) — compile-verified
//
#include <hip/hip_runtime.h>
#include <stdint.h>

#define B_ 128
#define T_ 256
#define E_ 256
#define H_ 512

typedef __attribute__((ext_vector_type(2))) float v2f;
typedef __attribute__((ext_vector_type(4))) float v4f;
typedef __attribute__((ext_vector_type(8))) float v8f;

// ---------- WMMA f32 16x16x4 fragment loads (CDNA5 ISA 7.12.2, wave32) ----------
// A: 16x4 f32 tile from row-major A (lda elems). Lanes 0-15: M=lane, K={k0,k0+1};
// lanes 16-31: M=lane-16, K={k0+2,k0+3}. One b64 load per lane.
static __device__ __forceinline__ v2f load_a4(const float* __restrict__ A,
                                              int lda, int row0, int k0) {
  const int lane = threadIdx.x & 31;
  const int m  = lane & 15;
  const int kb = (lane >> 4) << 1;               // 0 or 2
  return *(const v2f*)(A + (size_t)(row0 + m) * lda + (k0 + kb));
}
// B: 4x16 f32 tile; source pre-transposed to [N][K] row-major (ldb = K stride).
// Lanes 0-15: N=lane, K={k0,k0+1}; lanes 16-31: N=lane-16, K={k0+2,k0+3}.
static __device__ __forceinline__ v2f load_b4(const float* __restrict__ BT,
                                              int ldb, int n0, int k0) {
  const int lane = threadIdx.x & 31;
  const int n  = lane & 15;
  const int kb = (lane >> 4) << 1;
  return *(const v2f*)(BT + (size_t)(n0 + n) * ldb + (k0 + kb));
}

template <int K>
static __device__ __forceinline__ v8f gemm_tile(const float* __restrict__ A, int lda,
                                                const float* __restrict__ BT, int ldb,
                                                int row0, int n0) {
  v8f acc = {};
#pragma unroll 8
  for (int k0 = 0; k0 < K; k0 += 4) {
    v2f a = load_a4(A, lda, row0, k0);
    v2f b = load_b4(BT, ldb, n0, k0);
    acc = __builtin_amdgcn_wmma_f32_16x16x4_f32(false, a, false, b,
                                                (short)0, acc, false, false);
  }
  return acc;
}

// ---------- prep kernels ----------
// in: [K][N] f32 row-major -> out: [N][K] f32 row-major (weights become N-major for B-frags)
__global__ void k_transpose_f32(const float* __restrict__ in, float* __restrict__ out,
                                int K, int N) {
  int total = K * N;
  for (int i = blockIdx.x * blockDim.x + threadIdx.x; i < total; i += gridDim.x * blockDim.x) {
    int n = i / K, k = i % K;
    out[i] = in[(size_t)k * N + n];
  }
}
__global__ void k_init(const float* __restrict__ init_h, const float* __restrict__ init_c,
                       const float* __restrict__ init_ctx, const float* __restrict__ Wlstm,
                       float* __restrict__ hc0, float* __restrict__ ctx,
                       float* __restrict__ wl_last) {
  int i = blockIdx.x * blockDim.x + threadIdx.x;
  if (i < B_ * H_) {
    int b = i / H_, h = i % H_;
    hc0[(size_t)b * (2 * H_) + h]      = init_h[i];   // h half
    hc0[(size_t)b * (2 * H_) + H_ + h] = init_c[i];   // c half
  }
  if (i < B_ * E_) ctx[i] = init_ctx[i];
  if (i < H_)      wl_last[i] = Wlstm[(size_t)H_ * H_ + i];
}

// ---------- u_proj = enc @ Ua + bu  ([B*T,E] x [E,T] -> f32 [B*T,T]) ----------
__global__ void k_uproj(const float* __restrict__ enc, const float* __restrict__ UaT,
                        const float* __restrict__ bu, float* __restrict__ uproj) {
  int wave = blockIdx.x * (blockDim.x >> 5) + (threadIdx.x >> 5);
  int mt = wave >> 4, nt = wave & 15;              // 2048 x 16 tiles
  int row0 = mt * 16, n0 = nt * 16;
  v8f acc = gemm_tile<E_>(enc, E_, UaT, E_, row0, n0);
  const int lane = threadIdx.x & 31;
  const int col = lane & 15, rbase = (lane >> 4) << 3;
#pragma unroll
  for (int r = 0; r < 8; ++r) {
    int row = row0 + rbase + r, c = n0 + col;
    uproj[(size_t)row * T_ + c] = acc[r] + bu[c];
  }
}

// ---------- per-step: wa = [h|c] @ Wa + ba  ([B,2H] x [2H,T]) ----------
__global__ void k_wa(const float* __restrict__ hc, const float* __restrict__ WaT,
                     const float* __restrict__ ba, float* __restrict__ wa) {
  int wave = blockIdx.x * (blockDim.x >> 5) + (threadIdx.x >> 5);
  int mt = wave >> 4, nt = wave & 15;              // 8 x 16 tiles
  int row0 = mt * 16, n0 = nt * 16;
  v8f acc = gemm_tile<2 * H_>(hc, 2 * H_, WaT, 2 * H_, row0, n0);
  const int lane = threadIdx.x & 31;
  const int col = lane & 15, rbase = (lane >> 4) << 3;
#pragma unroll
  for (int r = 0; r < 8; ++r) {
    int row = row0 + rbase + r, c = n0 + col;
    wa[(size_t)row * T_ + c] = acc[r] + ba[c];
  }
}

// ---------- per-step: score/tanh/softmax/ctx/y_tilde, one block per batch row ----------
__global__ void k_attn(const float* __restrict__ enc, const float* __restrict__ uproj,
                       const float* __restrict__ wa, const float* __restrict__ Va,
                       const float* __restrict__ bv, const float* __restrict__ Wctx,
                       const float* __restrict__ bctx, float* __restrict__ ctx_out,
                       float* __restrict__ yt_out, int tstep) {
  __shared__ float s_wa[T_], s_va[T_], s_w[T_], s_red[256], s_ctx[E_];
  const int b = blockIdx.x, tid = threadIdx.x;
  s_wa[tid] = wa[(size_t)b * T_ + tid];
  s_va[tid] = Va[tid];
  __syncthreads();

  // score[tid] = sum_t' tanh(wa[t'] + u_proj[b,tid,t']) * Va[t']
  const float* up = uproj + ((size_t)b * T_ + tid) * T_;
  float acc = 0.f;
  for (int tp = 0; tp < T_; tp += 4) {
    if ((tp & 31) == 0 && tp + 128 < T_)
      __builtin_prefetch(up + tp + 128, 0, 1);     // global_prefetch_b8
    v4f u = *(const v4f*)(up + tp);
#pragma unroll
    for (int j = 0; j < 4; ++j)
      acc += tanhf(s_wa[tp + j] + u[j]) * s_va[tp + j];
  }
  float score = acc + bv[0];

  // softmax over the T positions (one per thread)
  s_red[tid] = score; __syncthreads();
  for (int s = 128; s > 0; s >>= 1) {
    if (tid < s) s_red[tid] = fmaxf(s_red[tid], s_red[tid + s]);
    __syncthreads();
  }
  float mx = s_red[0]; __syncthreads();
  float e = __expf(score - mx);
  s_red[tid] = e; __syncthreads();
  for (int s = 128; s > 0; s >>= 1) {
    if (tid < s) s_red[tid] += s_red[tid + s];
    __syncthreads();
  }
  float w = e / s_red[0];
  s_w[tid] = w;
  __syncthreads();

  // ctx[e] = sum_t w[t] * enc[b,t,e]  (coalesced across tid==e)
  float cacc = 0.f;
  for (int t = 0; t < T_; ++t) cacc += s_w[t] * enc[((size_t)b * T_ + t) * E_ + tid];
  s_ctx[tid] = cacc;
  ctx_out[(size_t)b * E_ + tid] = cacc;
  __syncthreads();

  // y_tilde = concat(x_t, ctx) . Wctx + bctx
  float part = enc[((size_t)b * T_ + tstep) * E_ + tid] * Wctx[tid]
             + s_ctx[tid] * Wctx[E_ + tid];
  s_red[tid] = part; __syncthreads();
  for (int s = 128; s > 0; s >>= 1) {
    if (tid < s) s_red[tid] += s_red[tid + s];
    __syncthreads();
  }
  if (tid == 0) yt_out[b] = s_red[0] + bctx[0];
}

// ---------- per-step: g = h @ Wlstm[:H] + yt*Wlstm[H] + b, fused LSTM update ----------
// hc layout: [B][0:H]=h, [B][H:2H]=c (fp32). Ping-pong in/out to avoid RAW across waves.
__global__ void k_gate(const float* __restrict__ hc_in, const float* __restrict__ WlT,
                       const float* __restrict__ wl_last, const float* __restrict__ blstm,
                       const float* __restrict__ yt, float* __restrict__ hc_out) {
  int wave = blockIdx.x * (blockDim.x >> 5) + (threadIdx.x >> 5);
  int mt = wave >> 5, nt = wave & 31;              // 8 x 32 tiles
  int row0 = mt * 16, n0 = nt * 16;
  v8f acc = gemm_tile<H_>(hc_in, 2 * H_, WlT, H_, row0, n0);  // A = h half
  const int lane = threadIdx.x & 31;
  const int col = lane & 15, rbase = (lane >> 4) << 3;
#pragma unroll
  for (int r = 0; r < 8; ++r) {
    int b = row0 + rbase + r, nh = n0 + col;
    float g  = acc[r] + yt[b] * wl_last[nh] + blstm[nh];
    float sg = 1.0f / (1.0f + __expf(-g));
    float tg = tanhf(g);
    float co = hc_in[(size_t)b * 2 * H_ + H_ + nh];
    float cn = sg * co + sg * tg;
    float hn = sg * tanhf(cn);
    hc_out[(size_t)b * 2 * H_ + nh]      = hn;
    hc_out[(size_t)b * 2 * H_ + H_ + nh] = cn;
  }
}

// ---------- head ----------
__global__ void k_hctx(const float* __restrict__ hc, const float* __restrict__ ctx,
                       float* __restrict__ hctx) {
  int i = blockIdx.x * blockDim.x + threadIdx.x;
  if (i >= B_ * (H_ + E_)) return;
  int b = i / (H_ + E_), j = i % (H_ + E_);
  hctx[i] = (j < H_) ? hc[(size_t)b * 2 * H_ + j] : ctx[(size_t)b * E_ + (j - H_)];
}
__global__ void k_pred1(const float* __restrict__ hctx, const float* __restrict__ Wp1T,
                        const float* __restrict__ bp1, float* __restrict__ pred1) {
  int wave = blockIdx.x * (blockDim.x >> 5) + (threadIdx.x >> 5);
  int mt = wave >> 5, nt = wave & 31;              // 8 x 32 tiles
  int row0 = mt * 16, n0 = nt * 16;
  v8f acc = gemm_tile<H_ + E_>(hctx, H_ + E_, Wp1T, H_ + E_, row0, n0);
  const int lane = threadIdx.x & 31;
  const int col = lane & 15, rbase = (lane >> 4) << 3;
#pragma unroll
  for (int r = 0; r < 8; ++r) {
    int b = row0 + rbase + r, nh = n0 + col;
    pred1[(size_t)b * H_ + nh] = acc[r] + bp1[nh];
  }
}
__global__ void k_pred2(const float* __restrict__ pred1, const float* __restrict__ Wp2,
                        const float* __restrict__ bp2, float* __restrict__ out) {
  int b = blockIdx.x * blockDim.x + threadIdx.x;
  if (b >= B_) return;
  float acc = 0.f;
  for (int j = 0; j < H_; ++j) acc += pred1[(size_t)b * H_ + j] * Wp2[j];
  out[b] = acc + bp2[0];
}

// ---------- host ----------
extern "C" void kernel_launch(void* const* d_in, const int* in_sizes, int n_in,
                              void* d_out, int out_size, void* d_ws, size_t ws_size,
                              hipStream_t stream) {
  const float* enc      = (const float*)d_in[0];
  const float* init_h   = (const float*)d_in[2];
  const float* init_c   = (const float*)d_in[3];
  const float* init_ctx = (const float*)d_in[4];
  const float* Wa    = (const float*)d_in[5];
  const float* ba    = (const float*)d_in[6];
  const float* Ua    = (const float*)d_in[7];
  const float* bu    = (const float*)d_in[8];
  const float* Va    = (const float*)d_in[9];
  const float* bv    = (const float*)d_in[10];
  const float* Wctx  = (const float*)d_in[11];
  const float* bctx  = (const float*)d_in[12];
  const float* Wlstm = (const float*)d_in[13];
  const float* blstm = (const float*)d_in[14];
  const float* Wp1   = (const float*)d_in[15];
  const float* bp1   = (const float*)d_in[16];
  const float* Wp2   = (const float*)d_in[17];
  const float* bp2   = (const float*)d_in[18];
  float* out = (float*)d_out;
  (void)in_sizes; (void)n_in; (void)out_size; (void)ws_size;

  size_t off = 0;
  auto alloc = [&](size_t bytes) -> void* {
    void* p = (char*)d_ws + off;
    off += (bytes + 255) & ~(size_t)255;
    return p;
  };
  float* uproj = (float*)alloc((size_t)B_ * T_ * T_ * 4);        // 33.5 MB (L2-resident)
  float* UaT   = (float*)alloc((size_t)T_ * E_ * 4);
  float* WaT   = (float*)alloc((size_t)T_ * 2 * H_ * 4);
  float* WlT   = (float*)alloc((size_t)H_ * H_ * 4);
  float* Wp1T  = (float*)alloc((size_t)H_ * (H_ + E_) * 4);
  float* hcA   = (float*)alloc((size_t)B_ * 2 * H_ * 4);
  float* hcB   = (float*)alloc((size_t)B_ * 2 * H_ * 4);
  float* hctx  = (float*)alloc((size_t)B_ * (H_ + E_) * 4);
  float* wl_last = (float*)alloc((size_t)H_ * 4);
  float* wa_buf  = (float*)alloc((size_t)B_ * T_ * 4);
  float* ctx     = (float*)alloc((size_t)B_ * E_ * 4);
  float* yt      = (float*)alloc((size_t)B_ * 4);
  float* pred1   = (float*)alloc((size_t)B_ * H_ * 4);

  // one-time prep: weight transposes to [N][K] (B-fragments become contiguous b64 loads)
  k_transpose_f32<<<256,  256, 0, stream>>>(Ua,    UaT,  E_,      T_);
  k_transpose_f32<<<1024, 256, 0, stream>>>(Wa,    WaT,  2 * H_,  T_);
  k_transpose_f32<<<1024, 256, 0, stream>>>(Wlstm, WlT,  H_,      H_);
  k_transpose_f32<<<1536, 256, 0, stream>>>(Wp1,   Wp1T, H_ + E_, H_);
  k_init<<<(B_ * H_ + 255) / 256, 256, 0, stream>>>(init_h, init_c, init_ctx, Wlstm,
                                                    hcA, ctx, wl_last);
  // step-invariant projection (fp32 WMMA)
  k_uproj<<<8192, 128, 0, stream>>>(enc, UaT, bu, uproj);

  // sequential scan (ping-pong fp32 h|c state; no precision loss in the recurrence)
  float* hc_cur = hcA;
  float* hc_nxt = hcB;
  for (int t = 0; t < T_; ++t) {
    k_wa  <<<32,  128, 0, stream>>>(hc_cur, WaT, ba, wa_buf);
    k_attn<<<B_,  256, 0, stream>>>(enc, uproj, wa_buf, Va, bv, Wctx, bctx, ctx, yt, t);
    k_gate<<<64,  128, 0, stream>>>(hc_cur, WlT, wl_last, blstm, yt, hc_nxt);
    float* tmp = hc_cur; hc_cur = hc_nxt; hc_nxt = tmp;
  }

  // prediction head
  k_hctx <<<(B_ * (H_ + E_) + 255) / 256, 256, 0, stream>>>(hc_cur, ctx, hctx);
  k_pred1<<<64, 128, 0, stream>>>(hctx, Wp1T, bp1, pred1);
  k_pred2<<<1,  128, 0, stream>>>(pred1, Wp2, bp2, out);
}